// VQVAE_3564822856193
// MI455X (gfx1250) — compile-verified
//
#include <hip/hip_runtime.h>
#include <math.h>

// ---------------------------------------------------------------------------
// VQ-VAE forward for MI455X (gfx1250, wave32).
// GEMMs: V_WMMA_F32_16X16X4_F32, weights staged in LDS by the Tensor Data
// Mover (TENSOR_LOAD_TO_LDS, double buffered, s_wait_tensorcnt), activations
// streamed from global/L2 (192MB L2 holds the whole working set).
// ---------------------------------------------------------------------------

typedef __attribute__((ext_vector_type(2))) float v2f;
typedef __attribute__((ext_vector_type(4))) float v4f;
typedef __attribute__((ext_vector_type(8))) float v8f;
typedef __attribute__((ext_vector_type(4))) unsigned int v4u;
typedef __attribute__((ext_vector_type(4))) int v4i;
typedef __attribute__((ext_vector_type(8))) int v8i;

#define INPUT_DIM   1568
#define NUM_LATENTS 98
#define EMBED_DIM   16
#define NUM_EMB     512
#define HIDDEN      256
#define BATCH       4096

#define VQ_ROWS     (BATCH * NUM_LATENTS)             // 401408
#define ZE_ELEMS    (BATCH * NUM_LATENTS * EMBED_DIM) // 6422528

// d_out layout (floats): [x_recon (6422528)] [vq_loss] [perplexity] [idx (401408)]
#define OUT_LOSS_OFF  ZE_ELEMS
#define OUT_PERP_OFF  (ZE_ELEMS + 1)
#define OUT_IDX_OFF   (ZE_ELEMS + 2)

#define KC 32   // K-chunk staged per TDM transfer (divides 1568 and 256)

// ---------------------------------------------------------------------------
// TDM: issue a 32(N-rows) x KC(K-cols) f32 tile load  W[n0.., kbase..] -> LDS.
// D# layout per CDNA5 ISA (2D tensor, groups 2/3 zero).
// ---------------------------------------------------------------------------
__device__ __forceinline__ void tdm_load_wtile(const float* Wt, int K, int n0,
                                               int kbase, unsigned int lds_off)
{
    unsigned long long ga = (unsigned long long)(uintptr_t)Wt +
        (((unsigned long long)n0 * (unsigned long long)K +
          (unsigned long long)kbase) << 2);

    v4u g0;
    g0.x = 1u;                                        // count=1 (valid user D#)
    g0.y = lds_off;                                   // lds_addr (bytes)
    g0.z = (unsigned int)ga;                          // global_addr[31:0]
    g0.w = ((unsigned int)(ga >> 32) & 0x01FFFFFFu)   // global_addr[56:32]
           | 0x80000000u;                             // type=2 ("image")

    v8i g1;
    g1[0] = (int)(2u << 16);          // wg_mask=0, data_size=2 (4 bytes)
    g1[1] = (int)((unsigned)KC << 16);// tensor_dim0[15:0] = KC
    g1[2] = (int)(32u << 16);         // tensor_dim0[31:16]=0, tensor_dim1[15:0]=32
    g1[3] = (int)((unsigned)KC << 16);// tensor_dim1[31:16]=0, tile_dim0 = KC
    g1[4] = 32;                       // tile_dim1 = 32, tile_dim2 = 0
    g1[5] = K;                        // tensor_dim0_stride[31:0] = K (elements)
    g1[6] = 0;                        // stride0[47:32]=0, stride1[15:0]=0
    g1[7] = 0;

    v4i gz = {0, 0, 0, 0};
#if __clang_major__ >= 23
    v8i gz8 = {0, 0, 0, 0, 0, 0, 0, 0};
    __builtin_amdgcn_tensor_load_to_lds(g0, g1, gz, gz, gz8, 0);
#else
    __builtin_amdgcn_tensor_load_to_lds(g0, g1, gz, gz, 0);
#endif
}

// ---------------------------------------------------------------------------
// C = act(A @ W^T + bias).  A:[M,K], W:[N,K], row-major.
// Block: 128 threads = 4 waves. Block tile: 128(M) x 32(N).
// Wave tile: 32x32 = 4 WMMA accumulators; A fragments from global (L2),
// B fragments from a TDM-staged, double-buffered LDS weight tile.
// ---------------------------------------------------------------------------
__global__ __launch_bounds__(128)
void gemm_bias_wmma(const float* __restrict__ A, const float* __restrict__ W,
                    const float* __restrict__ bias, float* __restrict__ C,
                    int M, int N, int K, int relu)
{
    __shared__ __align__(16) float wtile[2][32 * KC];   // 8 KB, double buffered

    const int lane = threadIdx.x & 31;
    const int wave = threadIdx.x >> 5;
    const int hh   = lane >> 4;     // K-pair select within fragment
    const int r    = lane & 15;

    const int m0 = (blockIdx.x * 4 + wave) * 32;
    const int n0 = blockIdx.y * 32;

    const float* arow0 = A + (size_t)(m0 + r)      * K + 2 * hh;
    const float* arow1 = A + (size_t)(m0 + 16 + r) * K + 2 * hh;

    const unsigned int lds0 = (unsigned int)(uintptr_t)&wtile[0][0];
    const unsigned int lds1 = (unsigned int)(uintptr_t)&wtile[1][0];

    v8f acc00 = {}, acc01 = {}, acc10 = {}, acc11 = {};

    const int NCHUNK = K / KC;
    if (wave == 0) tdm_load_wtile(W, K, n0, 0, lds0);   // prologue: chunk 0

    int cur = 0;
    for (int kc = 0; kc < NCHUNK; ++kc) {
        if (wave == 0) __builtin_amdgcn_s_wait_tensorcnt(0);
        __syncthreads();   // chunk `cur` resident; prior buffer fully consumed
        if (wave == 0 && kc + 1 < NCHUNK)
            tdm_load_wtile(W, K, n0, (kc + 1) * KC, cur ? lds0 : lds1);

        const int kbase = kc * KC;
        if (kc + 1 < NCHUNK) {  // prefetch next A chunk into L2/L0
            __builtin_prefetch(arow0 + kbase + KC, 0, 0);
            __builtin_prefetch(arow1 + kbase + KC, 0, 0);
        }

        const float* bt = &wtile[cur][0];
#pragma unroll
        for (int kk = 0; kk < KC; kk += 4) {
            v2f a0 = *(const v2f*)(arow0 + kbase + kk);
            v2f a1 = *(const v2f*)(arow1 + kbase + kk);
            v2f b0 = *(const v2f*)(bt + r        * KC + kk + 2 * hh);
            v2f b1 = *(const v2f*)(bt + (16 + r) * KC + kk + 2 * hh);
            acc00 = __builtin_amdgcn_wmma_f32_16x16x4_f32(false, a0, false, b0,
                                                          (short)0, acc00, false, false);
            acc01 = __builtin_amdgcn_wmma_f32_16x16x4_f32(false, a0, false, b1,
                                                          (short)0, acc01, false, false);
            acc10 = __builtin_amdgcn_wmma_f32_16x16x4_f32(false, a1, false, b0,
                                                          (short)0, acc10, false, false);
            acc11 = __builtin_amdgcn_wmma_f32_16x16x4_f32(false, a1, false, b1,
                                                          (short)0, acc11, false, false);
        }
        __syncthreads();
        cur ^= 1;
    }

    const int c0 = n0 + r;
    const int c1 = n0 + 16 + r;
    const float bi0 = bias[c0];
    const float bi1 = bias[c1];
#pragma unroll
    for (int v = 0; v < 8; ++v) {
        const int row0 = m0 + v + 8 * hh;
        const int row1 = row0 + 16;
        float x00 = acc00[v] + bi0, x01 = acc01[v] + bi1;
        float x10 = acc10[v] + bi0, x11 = acc11[v] + bi1;
        if (relu) {
            x00 = fmaxf(x00, 0.0f); x01 = fmaxf(x01, 0.0f);
            x10 = fmaxf(x10, 0.0f); x11 = fmaxf(x11, 0.0f);
        }
        C[(size_t)row0 * N + c0] = x00;
        C[(size_t)row0 * N + c1] = x01;
        C[(size_t)row1 * N + c0] = x10;
        C[(size_t)row1 * N + c1] = x11;
    }
}

// ---------------------------------------------------------------------------
// VQ nearest-codebook search. score = ||c||^2 - 2 z.c (||z||^2 constant/row).
// z.c is a WMMA GEMM against the LDS-staged codebook; per-lane running argmin
// over its n-residue class, then width-16 shfl_xor argmin (ties -> low index).
// ---------------------------------------------------------------------------
__global__ __launch_bounds__(256)
void vq_kernel(const float* __restrict__ z, const float* __restrict__ codebook,
               float* __restrict__ q, float* __restrict__ idx_out,
               int* __restrict__ maxidx)
{
    __shared__ __align__(16) float cb[NUM_EMB * EMBED_DIM]; // 32 KB
    __shared__ float cn[NUM_EMB];                           // 2 KB

    for (int i = threadIdx.x; i < NUM_EMB * EMBED_DIM; i += 256)
        cb[i] = codebook[i];
    __syncthreads();
    for (int e = threadIdx.x; e < NUM_EMB; e += 256) {
        float s = 0.0f;
#pragma unroll
        for (int t = 0; t < EMBED_DIM; ++t) { float c = cb[e * EMBED_DIM + t]; s += c * c; }
        cn[e] = s;
    }
    __syncthreads();

    const int lane = threadIdx.x & 31;
    const int wave = threadIdx.x >> 5;
    const int hh   = lane >> 4;
    const int r    = lane & 15;
    const int m0   = (blockIdx.x * 8 + wave) * 16;  // flat row tile base

    v2f a[4];
    const float* zrow = z + (size_t)(m0 + r) * EMBED_DIM + 2 * hh;
#pragma unroll
    for (int kk = 0; kk < 4; ++kk) a[kk] = *(const v2f*)(zrow + 4 * kk);

    float best[8];
    int   bestI[8];
#pragma unroll
    for (int v = 0; v < 8; ++v) { best[v] = 3.4e38f; bestI[v] = 0x7fffffff; }

    for (int nt = 0; nt < NUM_EMB / 16; ++nt) {
        v8f acc = {};
#pragma unroll
        for (int kk = 0; kk < 4; ++kk) {
            v2f b = *(const v2f*)(&cb[(nt * 16 + r) * EMBED_DIM + 4 * kk + 2 * hh]);
            acc = __builtin_amdgcn_wmma_f32_16x16x4_f32(false, a[kk], false, b,
                                                        (short)0, acc, false, false);
        }
        const int   n   = nt * 16 + r;
        const float cnv = cn[n];
#pragma unroll
        for (int v = 0; v < 8; ++v) {
            float s = fmaf(-2.0f, acc[v], cnv);
            if (s < best[v] || (s == best[v] && n < bestI[v])) { best[v] = s; bestI[v] = n; }
        }
    }

#pragma unroll
    for (int v = 0; v < 8; ++v) {
#pragma unroll
        for (int mask = 8; mask >= 1; mask >>= 1) {
            float os = __shfl_xor(best[v],  mask, 16);
            int   oi = __shfl_xor(bestI[v], mask, 16);
            if (os < best[v] || (os == best[v] && oi < bestI[v])) { best[v] = os; bestI[v] = oi; }
        }
    }

    if (r == 0) {
#pragma unroll
        for (int v = 0; v < 8; ++v) {
            const int m  = m0 + v + 8 * hh;
            const int bi = bestI[v];
            idx_out[m] = (float)bi;
            atomicMax(&maxidx[m % NUM_LATENTS], bi);
            const v4f* src = (const v4f*)&cb[bi * EMBED_DIM];
            v4f* dst = (v4f*)&q[(size_t)m * EMBED_DIM];
            dst[0] = src[0]; dst[1] = src[1]; dst[2] = src[2]; dst[3] = src[3];
        }
    }
}

// ---------------------------------------------------------------------------
__global__ __launch_bounds__(256)
void loss_reduce(const float* __restrict__ z, const float* __restrict__ q,
                 float* __restrict__ accum, int n)
{
    float s = 0.0f;
    for (int i = blockIdx.x * 256 + threadIdx.x; i < n; i += gridDim.x * 256) {
        float d = q[i] - z[i];
        s = fmaf(d, d, s);
    }
#pragma unroll
    for (int m = 16; m >= 1; m >>= 1) s += __shfl_xor(s, m, 32);
    __shared__ float wsum[8];
    if ((threadIdx.x & 31) == 0) wsum[threadIdx.x >> 5] = s;
    __syncthreads();
    if (threadIdx.x == 0) {
        float t = 0.0f;
#pragma unroll
        for (int w = 0; w < 8; ++w) t += wsum[w];
        atomicAdd(accum, t);
    }
}

__global__ __launch_bounds__(128)
void finalize(const int* __restrict__ maxidx, const float* __restrict__ accum,
              float* __restrict__ out_loss, float* __restrict__ out_perp)
{
    __shared__ float ps[128];
    const int t = threadIdx.x;
    float p = 0.0f;
    if (t < NUM_LATENTS) {
        float a = 1.0f / ((float)maxidx[t] + 1.0f);
        p = __expf(-a * __logf(a + 1e-10f));
    }
    ps[t] = p;
    __syncthreads();
    if (t == 0) {
        float s = 0.0f;
        for (int i = 0; i < NUM_LATENTS; ++i) s += ps[i];
        *out_perp = s / (float)NUM_LATENTS;
        // forward value: q_latent == e_latent == MSE -> vq_loss = 1.25 * MSE
        *out_loss = 1.25f * (*accum) / (float)ZE_ELEMS;
    }
}

__global__ __launch_bounds__(128)
void init_ws(float* __restrict__ accum, int* __restrict__ maxidx)
{
    const int t = threadIdx.x;
    if (t == 0) *accum = 0.0f;
    if (t < NUM_LATENTS) maxidx[t] = 0;
}

// ---------------------------------------------------------------------------
extern "C" void kernel_launch(void* const* d_in, const int* in_sizes, int n_in,
                              void* d_out, int out_size, void* d_ws, size_t ws_size,
                              hipStream_t stream)
{
    (void)in_sizes; (void)n_in; (void)out_size; (void)ws_size;

    const float* x        = (const float*)d_in[0];
    const float* We1      = (const float*)d_in[1];
    const float* be1      = (const float*)d_in[2];
    const float* We2      = (const float*)d_in[3];
    const float* be2      = (const float*)d_in[4];
    const float* codebook = (const float*)d_in[5];
    const float* Wd1      = (const float*)d_in[6];
    const float* bd1      = (const float*)d_in[7];
    const float* Wd2      = (const float*)d_in[8];
    const float* bd2      = (const float*)d_in[9];

    float* out = (float*)d_out;

    float* ws    = (float*)d_ws;
    float* h     = ws;                              // [4096,256], reused for h2
    float* z     = h + (size_t)BATCH * HIDDEN;      // [4096,1568] = z_e
    float* q     = z + (size_t)ZE_ELEMS;            // [4096,1568] quantized
    float* accum = q + (size_t)ZE_ELEMS;            // 1 float
    int*   maxi  = (int*)(accum + 1);               // 98 ints

    init_ws<<<1, 128, 0, stream>>>(accum, maxi);

    // encoder
    gemm_bias_wmma<<<dim3(BATCH / 128, HIDDEN / 32), 128, 0, stream>>>(
        x, We1, be1, h, BATCH, HIDDEN, INPUT_DIM, 1);
    gemm_bias_wmma<<<dim3(BATCH / 128, (NUM_LATENTS * EMBED_DIM) / 32), 128, 0, stream>>>(
        h, We2, be2, z, BATCH, NUM_LATENTS * EMBED_DIM, HIDDEN, 0);

    // vector quantization: 401408 rows, 16 rows/wave, 8 waves/block
    vq_kernel<<<VQ_ROWS / 128, 256, 0, stream>>>(z, codebook, q,
                                                 out + OUT_IDX_OFF, maxi);

    // losses + perplexity
    loss_reduce<<<1024, 256, 0, stream>>>(z, q, accum, ZE_ELEMS);
    finalize<<<1, 128, 0, stream>>>(maxi, accum,
                                    out + OUT_LOSS_OFF, out + OUT_PERP_OFF);

    // decoder (straight-through forward value is q)
    gemm_bias_wmma<<<dim3(BATCH / 128, HIDDEN / 32), 128, 0, stream>>>(
        q, Wd1, bd1, h, BATCH, HIDDEN, NUM_LATENTS * EMBED_DIM, 1);
    gemm_bias_wmma<<<dim3(BATCH / 128, INPUT_DIM / 32), 128, 0, stream>>>(
        h, Wd2, bd2, out, BATCH, INPUT_DIM, HIDDEN, 0);
}